// ProLoNet_37718402794064
// MI455X (gfx1250) — compile-verified
//
#include <hip/hip_runtime.h>
#include <math.h>

typedef __attribute__((ext_vector_type(2))) float v2f;
typedef __attribute__((ext_vector_type(8))) float v8f;

#define DEPTH      12
#define N_NODES    4095      // internal nodes, heap order
#define N_LEAVES   4096
#define INPUT_DIM  256
#define N_ACTIONS  10

// ---------------------------------------------------------------------------
// Kernel A: per-node gates via V_WMMA_F32_16X16X4_F32.
//   y[n] = dot(x, W[n,:]);  s[n] = sigmoid((y[n] - c[n]) * alpha)
// One wave handles one 16-node tile (B-matrix columns = nodes), x is
// broadcast across the 16 M-rows of A; K runs 256/4 = 64 WMMA steps.
// ---------------------------------------------------------------------------
__global__ __launch_bounds__(256) void prolonet_gates_wmma(
    const float* __restrict__ x,      // [256]
    const float* __restrict__ W,      // [4095,256] row-major
    const float* __restrict__ c,      // [4095]
    const float* __restrict__ alpha,  // [1]
    float* __restrict__ s_out)        // [4095] gate values
{
    const int lane      = threadIdx.x & 31;
    const int wave      = (blockIdx.x * blockDim.x + threadIdx.x) >> 5;
    const int node_base = wave * 16;             // 256 waves -> tiles 0..255
    const int half      = lane >> 4;             // K-group: 0 -> K{0,1}, 1 -> K{2,3}
    const int col       = lane & 15;             // N (node within tile) / M index

    const int row  = node_base + col;
    const int rowc = row < (N_NODES - 1) ? row : (N_NODES - 1);  // clamp pad node

    const float* wrow = W + (size_t)rowc * INPUT_DIM + half * 2;
    const float* xseg = x + half * 2;

    v8f acc = {};
#pragma unroll 8
    for (int k0 = 0; k0 < INPUT_DIM; k0 += 4) {
        v2f a = *(const v2f*)(xseg + k0);   // A[m,k] = x[k0+k]  (all m)
        v2f b = *(const v2f*)(wrow + k0);   // B[k,n] = W[node_base+n, k0+k]
        // (neg_a, A, neg_b, B, c_mod, C, reuse_a, reuse_b)
        acc = __builtin_amdgcn_wmma_f32_16x16x4_f32(
            false, a, false, b, (short)0, acc, false, false);
    }

    // Every M-row of D is identical; acc[0] on lanes 0..15 = y[node_base+lane].
    if (lane < 16 && row < N_NODES) {
        float z = (acc[0] - c[row]) * alpha[0];
        s_out[row] = 1.0f / (1.0f + __expf(-z));
    }
}

// ---------------------------------------------------------------------------
// Kernel B: leaf path products (heap traversal), probs @ action_probs,
// deterministic LDS tree reduction, softmax. Single workgroup.
// ---------------------------------------------------------------------------
__global__ __launch_bounds__(512) void prolonet_leaves_softmax(
    const float* __restrict__ s_buf,  // [4095]
    const float* __restrict__ ap,     // [4096,10]
    float* __restrict__ out)          // [10]
{
    __shared__ float s_lds[N_NODES];              // 16380 B
    __shared__ float part[512][N_ACTIONS];        // 20480 B (of 320 KB WGP LDS)

    const int tid = threadIdx.x;

    for (int i = tid; i < N_NODES; i += 512) s_lds[i] = s_buf[i];
    __syncthreads();

    float act[N_ACTIONS];
#pragma unroll
    for (int a = 0; a < N_ACTIONS; ++a) act[a] = 0.0f;

#pragma unroll
    for (int j = 0; j < N_LEAVES / 512; ++j) {
        const int leaf = tid + j * 512;
        float p  = 1.0f;
        int node = 0;
#pragma unroll
        for (int d = DEPTH - 1; d >= 0; --d) {
            const int bit = (leaf >> d) & 1;
            const float g = s_lds[node];
            p   *= bit ? (1.0f - g) : g;        // right uses (1-s), left uses s
            node = (node << 1) + 1 + bit;       // heap children 2k+1 / 2k+2
        }
        const float* apr = ap + (size_t)leaf * N_ACTIONS;
#pragma unroll
        for (int a = 0; a < N_ACTIONS; ++a) act[a] = fmaf(p, apr[a], act[a]);
    }

#pragma unroll
    for (int a = 0; a < N_ACTIONS; ++a) part[tid][a] = act[a];
    __syncthreads();

    // Deterministic tree reduction over 512 partials.
    for (int stride = 256; stride > 0; stride >>= 1) {
        if (tid < stride) {
#pragma unroll
            for (int a = 0; a < N_ACTIONS; ++a)
                part[tid][a] += part[tid + stride][a];
        }
        __syncthreads();
    }

    if (tid == 0) {
        float m = part[0][0];
        for (int a = 1; a < N_ACTIONS; ++a) m = fmaxf(m, part[0][a]);
        float e[N_ACTIONS];
        float sum = 0.0f;
        for (int a = 0; a < N_ACTIONS; ++a) { e[a] = __expf(part[0][a] - m); sum += e[a]; }
        const float inv = 1.0f / sum;
        for (int a = 0; a < N_ACTIONS; ++a) out[a] = e[a] * inv;
    }
}

// ---------------------------------------------------------------------------
extern "C" void kernel_launch(void* const* d_in, const int* in_sizes, int n_in,
                              void* d_out, int out_size, void* d_ws, size_t ws_size,
                              hipStream_t stream) {
    (void)in_sizes; (void)n_in; (void)out_size; (void)ws_size;
    const float* x     = (const float*)d_in[0];  // input_data   [1,256]
    const float* W     = (const float*)d_in[1];  // weights      [4095,256]
    const float* c     = (const float*)d_in[2];  // comparators  [4095]
    const float* alpha = (const float*)d_in[3];  // alpha        [1]
    const float* ap    = (const float*)d_in[4];  // action_probs [4096,10]
    // d_in[5], d_in[6] (path sig masks) are implied by the heap traversal.

    float* s_buf = (float*)d_ws;                 // 4095 gate values

    // 256 tiles of 16 nodes, one per wave: 32 blocks x 8 waves.
    prolonet_gates_wmma<<<32, 256, 0, stream>>>(x, W, c, alpha, s_buf);
    prolonet_leaves_softmax<<<1, 512, 0, stream>>>(s_buf, ap, (float*)d_out);
}